// MGA_51685636440337
// MI455X (gfx1250) — compile-verified
//
#include <hip/hip_runtime.h>
#include <math.h>

#define B_ 32
#define C_ 256
#define N_ 4096   // H*W = 64*64

typedef float v2f __attribute__((ext_vector_type(2)));
typedef float v8f __attribute__((ext_vector_type(8)));

// ---------------------------------------------------------------------------
// Kernel A: per-batch gating maps.  m = sigmoid(prior); mask = [m, 1-m]
// key  = Wk*mask + bk  -> k0,k1 [B,N]
// value= Wv*mask + bv  -> v0,v1 [B,N]
// ksum[b][k] = sum_n key_k[b][n]   (needed for the bq term of energy)
// ---------------------------------------------------------------------------
__global__ void mga_maps(const float* __restrict__ prior,
                         const float* __restrict__ Wk, const float* __restrict__ bk,
                         const float* __restrict__ Wv, const float* __restrict__ bv,
                         float* __restrict__ k0, float* __restrict__ k1,
                         float* __restrict__ v0, float* __restrict__ v1,
                         float* __restrict__ ksum) {
    __shared__ float r0[256], r1[256];
    const int b = blockIdx.x, t = threadIdx.x;
    const float wk00 = Wk[0], wk01 = Wk[1], wk10 = Wk[2], wk11 = Wk[3];
    const float wv00 = Wv[0], wv01 = Wv[1], wv10 = Wv[2], wv11 = Wv[3];
    const float bk0 = bk[0], bk1 = bk[1], bv0 = bv[0], bv1 = bv[1];
    const float* p = prior + (size_t)b * N_;
    float* K0 = k0 + (size_t)b * N_; float* K1 = k1 + (size_t)b * N_;
    float* V0 = v0 + (size_t)b * N_; float* V1 = v1 + (size_t)b * N_;
    float s0 = 0.f, s1 = 0.f;
    for (int n = t; n < N_; n += 256) {
        float m  = 1.0f / (1.0f + expf(-p[n]));
        float mc = 1.0f - m;
        float a = wk00 * m + wk01 * mc + bk0;
        float c = wk10 * m + wk11 * mc + bk1;
        K0[n] = a;  K1[n] = c;
        V0[n] = wv00 * m + wv01 * mc + bv0;
        V1[n] = wv10 * m + wv11 * mc + bv1;
        s0 += a;  s1 += c;
    }
    r0[t] = s0; r1[t] = s1; __syncthreads();
    for (int s = 128; s > 0; s >>= 1) {
        if (t < s) { r0[t] += r0[t + s]; r1[t] += r1[t + s]; }
        __syncthreads();
    }
    if (t == 0) { ksum[b * 2 + 0] = r0[0]; ksum[b * 2 + 1] = r1[0]; }
}

// ---------------------------------------------------------------------------
// Kernel B: S[b,c,k] = sum_n x[b,c,n] * key_k[b,n].  One block handles one
// batch and 8 consecutive channels; key maps cached in LDS (32 KB).
// This is the dominant HBM stream (reads all of x once, ~128 MiB).
// ---------------------------------------------------------------------------
__global__ void mga_skc(const float* __restrict__ x,
                        const float* __restrict__ k0, const float* __restrict__ k1,
                        float* __restrict__ S) {
    __shared__ float4 k0s[N_ / 4];   // 16 KB
    __shared__ float4 k1s[N_ / 4];   // 16 KB
    __shared__ float r0[256], r1[256];
    const int b  = blockIdx.x >> 5;      // 32 channel-groups per batch
    const int cg = blockIdx.x & 31;
    const int t  = threadIdx.x;
    const float4* K0 = (const float4*)(k0 + (size_t)b * N_);
    const float4* K1 = (const float4*)(k1 + (size_t)b * N_);
    for (int i = t; i < N_ / 4; i += 256) { k0s[i] = K0[i]; k1s[i] = K1[i]; }
    __syncthreads();
    for (int cc = 0; cc < 8; ++cc) {
        const int c = cg * 8 + cc;
        const float4* xr = (const float4*)(x + ((size_t)b * C_ + c) * N_);
        float a0 = 0.f, a1 = 0.f;
        for (int i = t; i < N_ / 4; i += 256) {
            float4 xv = xr[i], w0 = k0s[i], w1 = k1s[i];
            a0 += xv.x * w0.x + xv.y * w0.y + xv.z * w0.z + xv.w * w0.w;
            a1 += xv.x * w1.x + xv.y * w1.y + xv.z * w1.z + xv.w * w1.w;
        }
        r0[t] = a0; r1[t] = a1; __syncthreads();
        for (int s = 128; s > 0; s >>= 1) {
            if (t < s) { r0[t] += r0[t + s]; r1[t] += r1[t + s]; }
            __syncthreads();
        }
        if (t == 0) {
            S[((size_t)b * C_ + c) * 2 + 0] = r0[0];
            S[((size_t)b * C_ + c) * 2 + 1] = r1[0];
        }
        __syncthreads();
    }
}

// ---------------------------------------------------------------------------
// Kernel C: energy tile via V_WMMA_F32_16X16X4_F32, then 2-way softmax.
// One wave (32 lanes) per (batch, 16-row tile):  E[16,16] = Wq[16,256]·B
// where B[k, n] holds S columns in n=0,1 and zeros elsewhere.
// A slot (half h, vgpr v) <-> K = v + 2h  (ISA 16x4 f32 table); B mirrored.
// D layout: vgpr r, lane l -> M = r + 8*(l>=16), N = l&15.
// B operand built branch-free: loop-invariant lane masks m0/m1 turn the
// column-select into FMAs (no exec manipulation inside the WMMA loop).
// ---------------------------------------------------------------------------
__global__ void mga_attn(const float* __restrict__ Wq, const float* __restrict__ bq,
                         const float* __restrict__ S, const float* __restrict__ ksum,
                         float* __restrict__ att) {
    const int b    = blockIdx.x >> 4;
    const int tile = blockIdx.x & 15;
    const int lane = threadIdx.x;
    const int h = lane >> 4;       // lane half
    const int j = lane & 15;       // N column (also A row index)
    __shared__ float El[16][2];

    const float m0 = (j == 0) ? 1.0f : 0.0f;   // loop-invariant column masks
    const float m1 = (j == 1) ? 1.0f : 0.0f;

    v8f acc = {0.f, 0.f, 0.f, 0.f, 0.f, 0.f, 0.f, 0.f};
    const float* wrow = Wq + (size_t)(tile * 16 + j) * C_;   // A row M = lane&15
    const float* Sb   = S + (size_t)b * C_ * 2;

    for (int kk = 0; kk < C_; kk += 4) {
        const int k2 = kk + 2 * h;                  // K slots: v + 2h
        v2f a = *(const v2f*)(wrow + k2);           // A: Wq[row][k2], Wq[row][k2+1]
        float4 sv = *(const float4*)(Sb + k2 * 2);  // S[k2][0..1], S[k2+1][0..1]
        v2f bvv;
        bvv.x = sv.x * m0 + sv.y * m1;              // B[k2  ][j]
        bvv.y = sv.z * m0 + sv.w * m1;              // B[k2+1][j]
        acc = __builtin_amdgcn_wmma_f32_16x16x4_f32(
            /*neg_a=*/false, a, /*neg_b=*/false, bvv,
            /*c_mod=*/(short)0, acc, /*reuse_a=*/false, /*reuse_b=*/false);
    }

    if (j < 2) {
#pragma unroll
        for (int r = 0; r < 8; ++r) El[r + 8 * h][j] = acc[r];
    }
    __syncthreads();
    if (lane < 16) {
        const int c = tile * 16 + lane;
        // energy / sqrt(N) with N=4096 -> *1/64, then softmax over k (size 2)
        float e0 = (El[lane][0] + bq[c] * ksum[b * 2 + 0]) * 0.015625f;
        float e1 = (El[lane][1] + bq[c] * ksum[b * 2 + 1]) * 0.015625f;
        float mx = fmaxf(e0, e1);
        float p0 = expf(e0 - mx), p1 = expf(e1 - mx);
        float inv = 1.0f / (p0 + p1);
        att[((size_t)b * C_ + c) * 2 + 0] = p0 * inv;
        att[((size_t)b * C_ + c) * 2 + 1] = p1 * inv;
    }
}

// ---------------------------------------------------------------------------
// Kernel D: out = gamma*(att0*v0 + att1*v1) + x, float4-vectorized.
// Each block lies inside one (b,c) row so att loads are uniform.
// ---------------------------------------------------------------------------
__global__ void mga_out(const float* __restrict__ x, const float* __restrict__ att,
                        const float* __restrict__ v0, const float* __restrict__ v1,
                        const float* __restrict__ gamma, float* __restrict__ out) {
    const size_t idx = (size_t)blockIdx.x * 256 + threadIdx.x;  // float4 index
    const int    n4  = (int)(idx & (N_ / 4 - 1));
    const size_t bc  = idx >> 10;            // N_/4 == 1024
    const int    b   = (int)(bc >> 8);       // C_ == 256
    const float g  = gamma[0];
    const float a0 = att[bc * 2 + 0] * g;
    const float a1 = att[bc * 2 + 1] * g;
    float4 xx = ((const float4*)x)[idx];
    float4 w0 = ((const float4*)(v0 + (size_t)b * N_))[n4];
    float4 w1 = ((const float4*)(v1 + (size_t)b * N_))[n4];
    float4 o;
    o.x = a0 * w0.x + a1 * w1.x + xx.x;
    o.y = a0 * w0.y + a1 * w1.y + xx.y;
    o.z = a0 * w0.z + a1 * w1.z + xx.z;
    o.w = a0 * w0.w + a1 * w1.w + xx.w;
    ((float4*)out)[idx] = o;
}

extern "C" void kernel_launch(void* const* d_in, const int* in_sizes, int n_in,
                              void* d_out, int out_size, void* d_ws, size_t ws_size,
                              hipStream_t stream) {
    const float* x     = (const float*)d_in[0];
    const float* prior = (const float*)d_in[1];
    const float* Wq    = (const float*)d_in[2];
    const float* bq    = (const float*)d_in[3];
    const float* Wk    = (const float*)d_in[4];
    const float* bk    = (const float*)d_in[5];
    const float* Wv    = (const float*)d_in[6];
    const float* bv    = (const float*)d_in[7];
    const float* gamma = (const float*)d_in[8];
    float* out = (float*)d_out;

    // workspace layout (floats): k0,k1,v0,v1 [B*N] each | ksum [B*2] | S [B*C*2] | att [B*C*2]
    float* ws = (float*)d_ws;
    const size_t BN = (size_t)B_ * N_;
    float* k0   = ws;
    float* k1   = ws + BN;
    float* v0   = ws + 2 * BN;
    float* v1   = ws + 3 * BN;
    float* ksum = ws + 4 * BN;
    float* S    = ksum + 2 * B_;
    float* att  = S + 2 * (size_t)B_ * C_;

    mga_maps<<<B_, 256, 0, stream>>>(prior, Wk, bk, Wv, bv, k0, k1, v0, v1, ksum);
    mga_skc <<<B_ * (C_ / 8), 256, 0, stream>>>(x, k0, k1, S);
    mga_attn<<<B_ * 16, 32, 0, stream>>>(Wq, bq, S, ksum, att);
    mga_out <<<(B_ * C_ * (N_ / 4)) / 256, 256, 0, stream>>>(x, att, v0, v1, gamma, out);
}